// KANLayer_16552803958790
// MI455X (gfx1250) — compile-verified
//
#include <hip/hip_runtime.h>
#include <hip/hip_bf16.h>

// KAN Chebyshev layer: out[N,8] = sum_{f,s} T_s(x[n,f]) * W[f,s,o]
// Batched skinny GEMM T[N x 128] * W[128 x 8] on the fp32 WMMA pipe
// (V_WMMA_F32_16X16X4_F32). Memory-bound: ~64 MB traffic -> ~2.75 us
// HBM floor on MI455X; fp32 WMMA keeps reference precision.
//
// Round-1 fix: the Chebyshev values are now 16 NAMED scalars (not a
// float[16] array) so the hi/lo half-wave selection compiles to
// v_cndmask instead of a dynamically-indexed alloca that PromoteAlloca
// was spilling through LDS (ds_store/ds_load + dscnt stalls).

typedef __attribute__((ext_vector_type(2))) float v2f;
typedef __attribute__((ext_vector_type(8))) float v8f;

#define IN_F 8
#define OUT_F 8
#define SPEC 16

__global__ __launch_bounds__(256) void kan_cheb_wmma(
    const float* __restrict__ x,    // (N, 8)
    const float* __restrict__ w,    // (8, 16, 8)
    float* __restrict__ out,        // (N, 8)
    int n)
{
    const int lane = threadIdx.x & 31;
    const int col  = lane & 15;         // N-column of B/D this lane owns
    const bool hi  = lane >= 16;        // high half-wave
    const int wavesPerBlock = blockDim.x >> 5;
    const int waveId = blockIdx.x * wavesPerBlock + (threadIdx.x >> 5);
    const int nWaves = gridDim.x * wavesPerBlock;
    const int nTiles = n >> 4;          // 16 samples per tile

    // ---- Preload B fragments (constant across all tiles) ----
    // B is 4x16 (K x N) per wmma step. 32-bit B layout (mirror of A):
    //   VGPR0: lanes 0-15 -> K=s0+0, lanes 16-31 -> K=s0+2
    //   VGPR1: lanes 0-15 -> K=s0+1, lanes 16-31 -> K=s0+3
    // K index = f*16 + s. Columns >= OUT_F are zero padding.
    v2f bf[IN_F][4];
    #pragma unroll
    for (int f = 0; f < IN_F; ++f) {
        #pragma unroll
        for (int kk = 0; kk < 4; ++kk) {
            const int s0 = kk * 4 + (hi ? 2 : 0);
            float b0 = 0.0f, b1 = 0.0f;
            if (col < OUT_F) {
                b0 = w[(f * SPEC + s0    ) * OUT_F + col];
                b1 = w[(f * SPEC + s0 + 1) * OUT_F + col];
            }
            bf[f][kk].x = b0;
            bf[f][kk].y = b1;
        }
    }

    // ---- Grid-stride over 16-sample tiles ----
    for (int t = waveId; t < nTiles; t += nWaves) {
        const int base = t << 4;
        // Lane m and lane 16+m both load sample base+m (same cacheline;
        // served by WGP$). Two 128-bit coalesced loads per lane.
        const float4* xp4 = (const float4*)(x + (size_t)(base + col) * IN_F);
        const float4 l4 = xp4[0];
        const float4 h4 = xp4[1];
        float xv[IN_F] = { l4.x, l4.y, l4.z, l4.w, h4.x, h4.y, h4.z, h4.w };

        v8f acc = {0.f, 0.f, 0.f, 0.f, 0.f, 0.f, 0.f, 0.f};

        #pragma unroll
        for (int f = 0; f < IN_F; ++f) {
            // Chebyshev recurrence in NAMED scalars: T0=1, T1=x,
            // Tk = 2x*T(k-1) - T(k-2). No array -> no alloca -> no LDS.
            const float xf = xv[f];
            const float x2 = xf + xf;
            const float T0  = 1.0f;
            const float T1  = xf;
            const float T2  = __builtin_fmaf(x2, T1,  -T0);
            const float T3  = __builtin_fmaf(x2, T2,  -T1);
            const float T4  = __builtin_fmaf(x2, T3,  -T2);
            const float T5  = __builtin_fmaf(x2, T4,  -T3);
            const float T6  = __builtin_fmaf(x2, T5,  -T4);
            const float T7  = __builtin_fmaf(x2, T6,  -T5);
            const float T8  = __builtin_fmaf(x2, T7,  -T6);
            const float T9  = __builtin_fmaf(x2, T8,  -T7);
            const float T10 = __builtin_fmaf(x2, T9,  -T8);
            const float T11 = __builtin_fmaf(x2, T10, -T9);
            const float T12 = __builtin_fmaf(x2, T11, -T10);
            const float T13 = __builtin_fmaf(x2, T12, -T11);
            const float T14 = __builtin_fmaf(x2, T13, -T12);
            const float T15 = __builtin_fmaf(x2, T14, -T13);

            // 4 wmma k-steps cover this feature's 16 spectral modes.
            // 32-bit A (16x4) layout:
            //   VGPR0: lanes 0-15 -> K=s0+0, lanes 16-31 -> K=s0+2
            //   VGPR1: lanes 0-15 -> K=s0+1, lanes 16-31 -> K=s0+3
            v2f a;
            a.x = hi ? T2 : T0;  a.y = hi ? T3 : T1;
            acc = __builtin_amdgcn_wmma_f32_16x16x4_f32(
                false, a, false, bf[f][0], (short)0, acc, false, false);
            a.x = hi ? T6 : T4;  a.y = hi ? T7 : T5;
            acc = __builtin_amdgcn_wmma_f32_16x16x4_f32(
                false, a, false, bf[f][1], (short)0, acc, false, false);
            a.x = hi ? T10 : T8;  a.y = hi ? T11 : T9;
            acc = __builtin_amdgcn_wmma_f32_16x16x4_f32(
                false, a, false, bf[f][2], (short)0, acc, false, false);
            a.x = hi ? T14 : T12;  a.y = hi ? T15 : T13;
            acc = __builtin_amdgcn_wmma_f32_16x16x4_f32(
                false, a, false, bf[f][3], (short)0, acc, false, false);
        }

        // ---- Store D (16x16 f32): VGPR r -> row r (lanes 0-15) or r+8
        // (lanes 16-31), column = col. Only columns < OUT_F are real.
        if (col < OUT_F) {
            float* op = out + (size_t)(base + (hi ? 8 : 0)) * OUT_F + col;
            #pragma unroll
            for (int r = 0; r < 8; ++r)
                op[(size_t)r * OUT_F] = acc[r];
        }
    }
}

extern "C" void kernel_launch(void* const* d_in, const int* in_sizes, int n_in,
                              void* d_out, int out_size, void* d_ws, size_t ws_size,
                              hipStream_t stream) {
    const float* x = (const float*)d_in[0];   // (N, 8) fp32
    const float* w = (const float*)d_in[1];   // (8, 16, 8) fp32
    float* out = (float*)d_out;               // (N, 8) fp32
    const int n = in_sizes[0] / IN_F;         // 1048576

    const int block = 256;                    // 8 waves
    const int grid = 1024;                    // 8192 waves -> 8 tiles/wave
    kan_cheb_wmma<<<grid, block, 0, stream>>>(x, w, out, n);
}